// NormmaxBisect_90924457656605
// MI455X (gfx1250) — compile-verified
//
#include <hip/hip_runtime.h>
#include <hip/hip_bf16.h>

#ifndef __has_builtin
#define __has_builtin(x) 0
#endif

#if __has_builtin(__builtin_amdgcn_global_load_async_to_lds_b128) && \
    __has_builtin(__builtin_amdgcn_s_wait_asynccnt)
#define NM_HAVE_ASYNC 1
#else
#define NM_HAVE_ASYNC 0
#endif

typedef float v4f __attribute__((ext_vector_type(4)));
typedef int   v4i __attribute__((ext_vector_type(4)));

#if NM_HAVE_ASYNC
typedef __attribute__((address_space(1))) v4i* nm_as1_v4i;  // global, 16B pointee
typedef __attribute__((address_space(3))) v4i* nm_as3_v4i;  // LDS, 16B pointee
#endif

#define NM_D     2048   // row length (last dim)
#define NM_CHUNK 16     // float4 chunks per lane: 16*4 floats * 32 lanes = 2048
#define NM_ITER  50     // bisection iterations (N_ITER)
#define NM_WAVES 8      // waves (= rows) per block

// Full-wave (32-lane) butterfly reductions; every lane ends with the result.
__device__ __forceinline__ float nm_wave_sum(float v) {
#pragma unroll
  for (int m = 16; m >= 1; m >>= 1) v += __shfl_xor(v, m, 32);
  return v;
}
__device__ __forceinline__ float nm_wave_max(float v) {
#pragma unroll
  for (int m = 16; m >= 1; m >>= 1) v = fmaxf(v, __shfl_xor(v, m, 32));
  return v;
}

// One wave32 per row of 2048 fp32. alpha = 1.5 -> 1/(alpha-1) = 2, so
// p = t^2 and p^alpha = t^3 with t = max(x - tau, 0): pure sub/max/mul/fma.
__global__ __launch_bounds__(NM_WAVES * 32) void
NormmaxBisect_90924457656605_kernel(const float* __restrict__ X,
                                    float* __restrict__ O, int nrows) {
  const int lane = threadIdx.x & 31;
  const int wv   = threadIdx.x >> 5;
  const int row  = blockIdx.x * NM_WAVES + wv;

#if NM_HAVE_ASYNC
  __shared__ float ldsbuf[NM_WAVES * NM_D];  // 64 KB: one 8 KB row slice per wave
#endif

  if (row >= nrows) return;

  const float* xrow = X + (size_t)row * NM_D;
  v4f xv[NM_CHUNK];  // 64 VGPRs of row data, resident for all 50 iterations

#if NM_HAVE_ASYNC
  {
    // Stage the row through LDS with the CDNA5 async copy engine, then pull
    // into registers. Each wave only touches its own slice -> no barrier,
    // just s_wait_asynccnt. Chunk layout: lane-contiguous 512B bursts.
    float* dst = ldsbuf + wv * NM_D;
#pragma unroll
    for (int i = 0; i < NM_CHUNK; ++i) {
      const float* g = xrow + i * 128 + lane * 4;
      float*       l = dst  + i * 128 + lane * 4;
      __builtin_amdgcn_global_load_async_to_lds_b128(
          (nm_as1_v4i)g, (nm_as3_v4i)l, /*offset=*/0, /*cpol=*/0);
    }
    __builtin_amdgcn_s_wait_asynccnt(0);
#pragma unroll
    for (int i = 0; i < NM_CHUNK; ++i)
      xv[i] = *(const v4f*)(dst + i * 128 + lane * 4);
  }
#else
  __builtin_prefetch(xrow + lane * 4, 0, 0);  // global_prefetch_b8
#pragma unroll
  for (int i = 0; i < NM_CHUNK; ++i)
    xv[i] = __builtin_nontemporal_load((const v4f*)(xrow + i * 128 + lane * 4));
#endif

  // Row max.
  float lmax = -3.402823466e38f;
#pragma unroll
  for (int i = 0; i < NM_CHUNK; ++i) {
    v4f x = xv[i];
    lmax = fmaxf(lmax, fmaxf(fmaxf(x.x, x.y), fmaxf(x.z, x.w)));
  }
  const float mx = nm_wave_max(lmax);

  // tau_lo = max - 1^(a-1);  dm0 = tau_hi - tau_lo = 1 - (1/d)^(a-1) = 1 - 1/sqrt(d)
  float tau_lo = mx - 1.0f;
  float dm     = 1.0f - rsqrtf((float)NM_D);
  float tau_m  = tau_lo + dm;

  for (int it = 0; it < NM_ITER; ++it) {
    dm *= 0.5f;
    tau_m = tau_lo + dm;
    float s = 0.0f;  // sum of t^3 over this lane's 64 elements
#pragma unroll
    for (int i = 0; i < NM_CHUNK; ++i) {
      v4f x = xv[i];
      float t0 = fmaxf(x.x - tau_m, 0.0f);
      float t1 = fmaxf(x.y - tau_m, 0.0f);
      float t2 = fmaxf(x.z - tau_m, 0.0f);
      float t3 = fmaxf(x.w - tau_m, 0.0f);
      s = fmaf(t0 * t0, t0, s);
      s = fmaf(t1 * t1, t1, s);
      s = fmaf(t2 * t2, t2, s);
      s = fmaf(t3 * t3, t3, s);
    }
    const float fsum = nm_wave_sum(s);   // wave-uniform
    if (fsum >= 1.0f) tau_lo = tau_m;    // f_m = fsum - 1 >= 0
  }

  // Final renormalization: p = t^2 with the *last* tau_m (as in the reference),
  // out = p / sum(p). Recompute t instead of burning 64 more VGPRs.
  float ps = 0.0f;
#pragma unroll
  for (int i = 0; i < NM_CHUNK; ++i) {
    v4f x = xv[i];
    float t0 = fmaxf(x.x - tau_m, 0.0f);
    float t1 = fmaxf(x.y - tau_m, 0.0f);
    float t2 = fmaxf(x.z - tau_m, 0.0f);
    float t3 = fmaxf(x.w - tau_m, 0.0f);
    ps = fmaf(t0, t0, ps);
    ps = fmaf(t1, t1, ps);
    ps = fmaf(t2, t2, ps);
    ps = fmaf(t3, t3, ps);
  }
  const float inv = 1.0f / nm_wave_sum(ps);

  float* orow = O + (size_t)row * NM_D;
#pragma unroll
  for (int i = 0; i < NM_CHUNK; ++i) {
    v4f x = xv[i];
    float t0 = fmaxf(x.x - tau_m, 0.0f);
    float t1 = fmaxf(x.y - tau_m, 0.0f);
    float t2 = fmaxf(x.z - tau_m, 0.0f);
    float t3 = fmaxf(x.w - tau_m, 0.0f);
    v4f o;
    o.x = t0 * t0 * inv;
    o.y = t1 * t1 * inv;
    o.z = t2 * t2 * inv;
    o.w = t3 * t3 * inv;
    __builtin_nontemporal_store(o, (v4f*)(orow + i * 128 + lane * 4));
  }
}

extern "C" void kernel_launch(void* const* d_in, const int* in_sizes, int n_in,
                              void* d_out, int out_size, void* d_ws, size_t ws_size,
                              hipStream_t stream) {
  (void)n_in; (void)d_ws; (void)ws_size; (void)out_size;
  const float* X = (const float*)d_in[0];
  float* O = (float*)d_out;
  const int nrows = in_sizes[0] / NM_D;  // 2*16*2048 = 65536 rows
  const int blocks = (nrows + NM_WAVES - 1) / NM_WAVES;
  NormmaxBisect_90924457656605_kernel<<<blocks, NM_WAVES * 32, 0, stream>>>(X, O, nrows);
}